// ModelNew_1949915152991
// MI455X (gfx1250) — compile-verified
//
#include <hip/hip_runtime.h>
#include <stdint.h>

#define NDIM 4096
#define BM 256          // block tile rows
#define BN 128          // block tile cols
#define BK 16           // K stage depth (double-buffered)
#define LDSS 20         // padded LDS stride (floats): 16B-aligned, conflict-free frag loads

#define A_OFF (BM * LDSS)        // floats per A buffer
#define B_BASE (2 * BM * LDSS)   // start of B buffers (floats)
#define B_OFF (BN * LDSS)        // floats per B buffer

typedef __attribute__((ext_vector_type(2))) float v2f;
typedef __attribute__((ext_vector_type(8))) float v8f;

// C = tril(tril(A) @ tril(B)), fp32, N = 4096.
// Block tile 256x128; 8 waves (wave32), each computes 64x64 (16 v8f accums)
// with V_WMMA_F32_16X16X4_F32. Interior K-stages are staged by the CDNA5
// async DMA path (global_load_async_to_lds_*, ASYNCcnt); diagonal-touching
// stages use a masked register path so tril zeros are exact.
__global__ __launch_bounds__(256) void tril_wmma_f32_kernel(const float* __restrict__ A,
                                                            const float* __restrict__ B,
                                                            float* __restrict__ C) {
  const int bj = blockIdx.x;               // block column (of 128)
  const int bi = blockIdx.y;               // block row (of 256)
  const int tid = (int)threadIdx.x;

  const int row0 = bi * BM;
  const int col0 = bj * BN;

  if (col0 > row0 + (BM - 1)) {
    // Entirely strict-upper block: write exact zeros (output buffer is poisoned).
    const float4 z = make_float4(0.f, 0.f, 0.f, 0.f);
#pragma unroll
    for (int i = 0; i < 32; ++i) {
      const int q = tid + i * 256;         // 256 rows x 32 float4
      const int r = q >> 5;
      const int c = (q & 31) * 4;
      *(float4*)(C + (size_t)(row0 + r) * NDIM + (col0 + c)) = z;
    }
    return;
  }

  // Ping-pong LDS: A[2][256x20], B[2][128x20] = 61440 bytes.
  __shared__ __align__(16) float smem[2 * BM * LDSS + 2 * BN * LDSS];

  const int wave = tid >> 5;
  const int lane = tid & 31;
  const int half = lane >> 4;              // 0: K pair {0,1}, 1: K pair {2,3}
  const int lq   = lane & 15;
  const int wm   = (wave & 3) * 64;        // wave M offset
  const int wn   = (wave >> 2) * 64;       // wave N offset

  v8f acc[4][4];
#pragma unroll
  for (int mt = 0; mt < 4; ++mt)
#pragma unroll
    for (int nt = 0; nt < 4; ++nt)
      acc[mt][nt] = (v8f){0.f, 0.f, 0.f, 0.f, 0.f, 0.f, 0.f, 0.f};

  const int kEnd = row0 + BM;              // live K band: [col0, row0+BM)
  const int S = (kEnd - col0) / BK;

  // Register staging (boundary stages only): A 256x16 -> 4 f4/thread, B 16x128 -> 2 f4/thread.
  float4 aR[4];
  float4 bR[2];

  // Issue staging for stage s into buffer buf (async DMA when no masking needed).
  auto issueStage = [&](int s, int buf) {
    const int k0 = col0 + s * BK;
    const bool asyA = (k0 + (BK - 1) <= row0);        // no k > row in tile
    const bool asyB = (k0 >= col0 + (BN - 1));        // no k < col in tile
    if (asyA) {
      const unsigned aBase = (unsigned)(uintptr_t)smem + (unsigned)(buf * A_OFF) * 4u;
#pragma unroll
      for (int i = 0; i < 4; ++i) {
        const int q  = (wave * 4 + i) * 32 + lane;    // [0,1024) f4 slots: 256 rows x 4 f4
        const int r  = q >> 2;
        const int kq = (q & 3) * 4;
        const unsigned la = aBase + (unsigned)(r * LDSS + kq) * 4u;
        const unsigned go = (unsigned)(((row0 + r) * NDIM + (k0 + kq)) * 4);
        asm volatile("global_load_async_to_lds_b128 %0, %1, %2"
                     :: "v"(la), "v"(go), "s"((const void*)A) : "memory");
      }
    } else {
#pragma unroll
      for (int i = 0; i < 4; ++i) {
        const int q  = tid + i * 256;
        const int r  = q >> 2;
        const int kq = (q & 3) * 4;
        aR[i] = *(const float4*)(A + (size_t)(row0 + r) * NDIM + (k0 + kq));
      }
    }
    if (asyB) {
      const unsigned bBase = (unsigned)(uintptr_t)smem + (unsigned)(B_BASE + buf * B_OFF) * 4u;
#pragma unroll
      for (int j = 0; j < 8; ++j) {                   // 64 wave-instrs x 32 dwords = 16x128
        const int g = wave * 8 + j;
        const int k = g >> 2;
        const int n = (g & 3) * 32 + lane;
        const unsigned lb = bBase + (unsigned)(n * LDSS + k) * 4u;  // transposed [n][k]
        const unsigned go = (unsigned)(((k0 + k) * NDIM + (col0 + n)) * 4);
        asm volatile("global_load_async_to_lds_b32 %0, %1, %2"
                     :: "v"(lb), "v"(go), "s"((const void*)B) : "memory");
      }
    } else {
#pragma unroll
      for (int i = 0; i < 2; ++i) {
        const int q  = tid + i * 256;
        const int k  = q >> 5;
        const int nq = (q & 31) * 4;
        bR[i] = *(const float4*)(B + (size_t)(k0 + k) * NDIM + (col0 + nq));
      }
    }
  };

  // Commit register-path staging (with exact tril masking) into buffer buf.
  auto finishStage = [&](int s, int buf) {
    const int k0 = col0 + s * BK;
    const bool asyA = (k0 + (BK - 1) <= row0);
    const bool asyB = (k0 >= col0 + (BN - 1));
    if (!asyA) {
      float* const la = smem + buf * A_OFF;
#pragma unroll
      for (int i = 0; i < 4; ++i) {
        const int q  = tid + i * 256;
        const int r  = q >> 2;
        const int kq = (q & 3) * 4;
        const int grow = row0 + r;
        const int gk   = k0 + kq;
        float4 v = aR[i];
        if (gk + 0 > grow) v.x = 0.f;
        if (gk + 1 > grow) v.y = 0.f;
        if (gk + 2 > grow) v.z = 0.f;
        if (gk + 3 > grow) v.w = 0.f;
        *(float4*)(la + r * LDSS + kq) = v;
      }
    }
    if (!asyB) {
      float* const lb = smem + B_BASE + buf * B_OFF;
#pragma unroll
      for (int i = 0; i < 2; ++i) {
        const int q  = tid + i * 256;
        const int k  = q >> 5;
        const int nq = (q & 31) * 4;
        const int gk = k0 + k;
        const float vv[4] = {bR[i].x, bR[i].y, bR[i].z, bR[i].w};
#pragma unroll
        for (int c = 0; c < 4; ++c)
          lb[(nq + c) * LDSS + k] = (gk < col0 + nq + c) ? 0.f : vv[c];
      }
    }
  };

  // Prologue: stage 0 into buffer 0.
  issueStage(0, 0);
  finishStage(0, 0);
  asm volatile("s_wait_asynccnt 0x0" ::: "memory");
  __syncthreads();

  for (int s = 0; s < S; ++s) {
    const int cur = s & 1;
    const bool haveNext = (s + 1 < S);
    if (haveNext) issueStage(s + 1, cur ^ 1);

    // ---- 4 K-steps; each: 8 fragment loads feed 16 fp32 WMMAs (64 per stage) ----
    const float* const la = smem + cur * A_OFF;
    const float* const lb = smem + B_BASE + cur * B_OFF;
#pragma unroll
    for (int kk = 0; kk < BK; kk += 4) {
      v2f afrag[4], bfrag[4];
#pragma unroll
      for (int mt = 0; mt < 4; ++mt)
        afrag[mt] = *(const v2f*)(la + (wm + mt * 16 + lq) * LDSS + kk + half * 2);
#pragma unroll
      for (int nt = 0; nt < 4; ++nt)
        bfrag[nt] = *(const v2f*)(lb + (wn + nt * 16 + lq) * LDSS + kk + half * 2);
#pragma unroll
      for (int mt = 0; mt < 4; ++mt)
#pragma unroll
        for (int nt = 0; nt < 4; ++nt)
          acc[mt][nt] = __builtin_amdgcn_wmma_f32_16x16x4_f32(
              false, afrag[mt], false, bfrag[nt],
              (short)0, acc[mt][nt], false, false);
    }

    if (haveNext) finishStage(s + 1, cur ^ 1);
    asm volatile("s_wait_asynccnt 0x0" ::: "memory");
    __syncthreads();
  }

  // ---- Store C: VGPR r holds M = half*8 + r, lane lq holds N ----
#pragma unroll
  for (int mt = 0; mt < 4; ++mt)
#pragma unroll
    for (int nt = 0; nt < 4; ++nt)
#pragma unroll
      for (int r = 0; r < 8; ++r) {
        const int row = row0 + wm + mt * 16 + half * 8 + r;
        const int col = col0 + wn + nt * 16 + lq;
        C[(size_t)row * NDIM + col] = acc[mt][nt][r];
      }
}

extern "C" void kernel_launch(void* const* d_in, const int* in_sizes, int n_in,
                              void* d_out, int out_size, void* d_ws, size_t ws_size,
                              hipStream_t stream) {
  (void)in_sizes; (void)n_in; (void)out_size; (void)d_ws; (void)ws_size;
  const float* A = (const float*)d_in[0];
  const float* B = (const float*)d_in[1];
  float* C = (float*)d_out;
  dim3 grid(NDIM / BN, NDIM / BM);  // (32, 16): x = block col, y = block row
  dim3 block(256);
  tril_wmma_f32_kernel<<<grid, block, 0, stream>>>(A, B, C);
}